// Heirachical_Loss_14594298872596
// MI455X (gfx1250) — compile-verified
//
#include <hip/hip_runtime.h>
#include <math.h>

// Hierarchical softmax loss, MI455X (gfx1250, wave32).
// outputs: [8192, 4096] fp32, target: [8192] i32.
// d_out (fp32): [0] = summed loss, [1..8192] = greedy-decoded leaf ids (as float).
// d_ws: 8192 fp32 per-row losses (deterministic 2-kernel reduction, no atomics).

#define NUM_LEAVES 4096
#define BATCH      8192
#define TPB        256
#define ROWS_PER_BLOCK 4
#define EPT        16            // 4096 / 256 elements per thread

// ---- CDNA5 async global->LDS path (guarded; fall back to sync copy) --------
#if defined(__HIP_DEVICE_COMPILE__) && \
    __has_builtin(__builtin_amdgcn_global_load_async_to_lds_b128) && \
    __has_builtin(__builtin_amdgcn_s_wait_asynccnt)
#define USE_ASYNC 1
typedef int v4i __attribute__((vector_size(16)));
#define WAIT_ASYNC(n) __builtin_amdgcn_s_wait_asynccnt(n)
#else
#define USE_ASYNC 0
#define WAIT_ASYNC(n)
#endif

__device__ __forceinline__ void stage_row(const float* __restrict__ g, float* l) {
  const int t = threadIdx.x;
#if USE_ASYNC
#pragma unroll
  for (int i = 0; i < 4; ++i) {
    const int off = (i * TPB + t) * 4;          // 16B granules, fully coalesced
    __builtin_amdgcn_global_load_async_to_lds_b128(
        (v4i*)(g + off), (v4i*)(l + off), 0, 0);
  }
#else
#pragma unroll
  for (int i = 0; i < 4; ++i) {
    const int off = (i * TPB + t) * 4;
    *reinterpret_cast<float4*>(l + off) = *reinterpret_cast<const float4*>(g + off);
  }
#endif
}

__global__ __launch_bounds__(TPB) void hier_loss_fwd(
    const float* __restrict__ outputs, const int* __restrict__ target,
    float* __restrict__ rowLoss, float* __restrict__ preds) {
  __shared__ float buf[2][NUM_LEAVES];   // double-buffered staged rows (2x16KB)
  __shared__ float lvl3[512];            // sums of 8 leaves
  __shared__ float lvl2[64];             // sums of 64 leaves
  __shared__ float lvl1[8];              // sums of 512 leaves
  __shared__ float red[8];               // per-wave max scratch (8 waves)

  const int t    = threadIdx.x;
  const int lane = t & 31;
  const int wid  = t >> 5;
  const int row0 = blockIdx.x * ROWS_PER_BLOCK;

  stage_row(outputs + (size_t)row0 * NUM_LEAVES, buf[0]);

#pragma unroll
  for (int i = 0; i < ROWS_PER_BLOCK; ++i) {
    const int row = row0 + i;
    if (i + 1 < ROWS_PER_BLOCK) {
      stage_row(outputs + (size_t)(row + 1) * NUM_LEAVES, buf[(i + 1) & 1]);
      WAIT_ASYNC(4);                     // loads complete in-order: buf[i&1] ready
    } else {
      WAIT_ASYNC(0);
    }
    __syncthreads();                     // all waves' async writes visible

    float* b = buf[i & 1];

    // ---- row max (wave32 shuffle tree + cross-wave via LDS) ----
    float v[EPT];
    float lmax = -3.402823466e38f;
#pragma unroll
    for (int k = 0; k < EPT; ++k) {
      v[k] = b[t * EPT + k];
      lmax = fmaxf(lmax, v[k]);
    }
#pragma unroll
    for (int m = 16; m >= 1; m >>= 1)
      lmax = fmaxf(lmax, __shfl_xor(lmax, m, 32));
    if (lane == 0) red[wid] = lmax;
    __syncthreads();
    float M = red[0];
#pragma unroll
    for (int w = 1; w < 8; ++w) M = fmaxf(M, red[w]);

    // ---- exp (in-place in LDS) + level-3 (blocks of 8) sums ----
    float s0 = 0.f, s1 = 0.f;
#pragma unroll
    for (int k = 0; k < 8; ++k) { float e = __expf(v[k] - M); b[t * EPT + k] = e; s0 += e; }
#pragma unroll
    for (int k = 8; k < 16; ++k) { float e = __expf(v[k] - M); b[t * EPT + k] = e; s1 += e; }
    lvl3[2 * t]     = s0;
    lvl3[2 * t + 1] = s1;
    __syncthreads();

    if (t < 64) {
      float s = 0.f;
#pragma unroll
      for (int k = 0; k < 8; ++k) s += lvl3[8 * t + k];
      lvl2[t] = s;
    }
    __syncthreads();
    if (t < 8) {
      float s = 0.f;
#pragma unroll
      for (int k = 0; k < 8; ++k) s += lvl2[8 * t + k];
      lvl1[t] = s;
    }
    __syncthreads();

    // ---- scalar tail: loss along target path + greedy decode ----
    if (t == 0) {
      float D = 0.f;
#pragma unroll
      for (int k = 0; k < 8; ++k) D += lvl1[k];
      const int tg = target[row];
      const float S4 = b[tg];
      const float S3 = lvl3[tg >> 3];
      const float S2 = lvl2[tg >> 6];
      const float S1 = lvl1[tg >> 9];
      // 2*(1-win) = 1 - (0.5*S1 + 0.25*S2 + 0.125*S3 + 0.125*S4)/D
      const float q = (0.5f * S1 + 0.25f * S2 + 0.125f * S3 + 0.125f * S4) / D;
      rowLoss[row] = logf(1.0f - q);

      int k1 = 0;
      { float best = lvl1[0];
        for (int j = 1; j < 8; ++j) if (lvl1[j] > best) { best = lvl1[j]; k1 = j; } }
      int k2 = 8 * k1;
      { float best = lvl2[k2]; int bi = 0;
        for (int j = 1; j < 8; ++j) if (lvl2[8 * k1 + j] > best) { best = lvl2[8 * k1 + j]; bi = j; }
        k2 = 8 * k1 + bi; }
      int k3 = 8 * k2;
      { float best = lvl3[k3]; int bi = 0;
        for (int j = 1; j < 8; ++j) if (lvl3[8 * k2 + j] > best) { best = lvl3[8 * k2 + j]; bi = j; }
        k3 = 8 * k2 + bi; }
      int k4 = 8 * k3;
      { float best = b[k4]; int bi = 0;
        for (int j = 1; j < 8; ++j) if (b[8 * k3 + j] > best) { best = b[8 * k3 + j]; bi = j; }
        k4 = 8 * k3 + bi; }
      preds[row] = (float)k4;
    }
    __syncthreads();                     // protect buf/lvl reuse next iteration
  }
}

__global__ __launch_bounds__(TPB) void reduce_loss(const float* __restrict__ rowLoss,
                                                   float* __restrict__ out) {
  __shared__ float red[8];
  const int t = threadIdx.x, lane = t & 31, wid = t >> 5;
  float s = 0.f;
  for (int j = t; j < BATCH; j += TPB) s += rowLoss[j];
#pragma unroll
  for (int m = 16; m >= 1; m >>= 1) s += __shfl_xor(s, m, 32);
  if (lane == 0) red[wid] = s;
  __syncthreads();
  if (t == 0) {
    float tot = 0.f;
#pragma unroll
    for (int w = 0; w < 8; ++w) tot += red[w];
    out[0] = tot;
  }
}

extern "C" void kernel_launch(void* const* d_in, const int* in_sizes, int n_in,
                              void* d_out, int out_size, void* d_ws, size_t ws_size,
                              hipStream_t stream) {
  (void)in_sizes; (void)n_in; (void)out_size; (void)ws_size;
  const float* outputs = (const float*)d_in[0];
  const int*   target  = (const int*)d_in[1];
  float* out     = (float*)d_out;
  float* rowLoss = (float*)d_ws;                 // 8192 fp32 scratch

  hier_loss_fwd<<<BATCH / ROWS_PER_BLOCK, TPB, 0, stream>>>(outputs, target, rowLoss, out + 1);
  reduce_loss<<<1, TPB, 0, stream>>>(rowLoss, out);
}